// GNNLayer_481036337942
// MI455X (gfx1250) — compile-verified
//
#include <hip/hip_runtime.h>

typedef __attribute__((ext_vector_type(2))) float v2f;
typedef __attribute__((ext_vector_type(8))) float v8f;

#define DIM 128  // DIM_IN == DIM_OUT == 128

// ---------------------------------------------------------------------------
// Kernel 0: zero the output accumulator (harness poisons d_out with 0xAA).
// ---------------------------------------------------------------------------
__global__ void gnn_zero_kernel(float4* __restrict__ out, int n4) {
    int i = blockIdx.x * blockDim.x + threadIdx.x;
    int stride = gridDim.x * blockDim.x;
    float4 z = make_float4(0.f, 0.f, 0.f, 0.f);
    for (; i < n4; i += stride) out[i] = z;
}

// ---------------------------------------------------------------------------
// Kernel 1: h = x @ W^T using V_WMMA_F32_16X16X4_F32 (fp32 matrix pipe).
//   x: [N, 128] row-major, W: [128, 128] row-major ([out_ch, in_ch])
//   h: [N, 128] row-major (workspace)
// One wave computes one 16x16 tile of h; 8 waves per block cover the 8
// column tiles of a 16-row strip. K=128 -> 32 WMMA steps of K=4.
//
// Fragment layouts (ISA 7.12.2, wave32):
//   A (16x4 f32):  lane L holds M=L%16, K = 2*(L/16) + {reg0,reg1}
//   B (4x16 f32):  lane L holds N=L%16, K = 2*(L/16) + {reg0,reg1}
//   C/D (16x16):   lane L, reg r holds M = r + 8*(L/16), N = L%16
// ---------------------------------------------------------------------------
__global__ void __launch_bounds__(256) gnn_gemm_wmma_kernel(
    const float* __restrict__ x, const float* __restrict__ W,
    float* __restrict__ h, int n_nodes) {
    const int lane  = threadIdx.x & 31;
    const int wave  = threadIdx.x >> 5;         // 0..7 -> output-column tile
    const int row0  = blockIdx.x * 16;          // 16 node rows per block
    const int col0  = wave * 16;                // 16 output channels per wave
    const int nm    = lane & 15;                // M for A, N for B/C/D
    const int khalf = (lane >> 4) * 2;          // 0 or 2

    // Clamp A-row so every lane issues a valid load (EXEC must stay all-1s
    // through the WMMA loop).
    int arow = row0 + nm;
    if (arow >= n_nodes) arow = n_nodes - 1;
    const float* __restrict__ xrow = x + (size_t)arow * DIM + khalf;
    // B[k][n] = W[col0+n][k]  (h[n][o] = sum_k x[n][k] * W[o][k])
    const float* __restrict__ wrow = W + (size_t)(col0 + nm) * DIM + khalf;

    v8f acc = {};
#pragma unroll 4
    for (int ks = 0; ks < DIM; ks += 4) {
        v2f a, b;
        a.x = xrow[ks + 0];
        a.y = xrow[ks + 1];
        b.x = wrow[ks + 0];
        b.y = wrow[ks + 1];
        // 8 args: (neg_a, A, neg_b, B, c_mod, C, reuse_a, reuse_b)
        acc = __builtin_amdgcn_wmma_f32_16x16x4_f32(
            false, a, false, b, (short)0, acc, false, false);
    }

    const int mhi = (lane >> 4) * 8;
    float* __restrict__ hbase = h + (size_t)row0 * DIM + col0;
    if (row0 + 16 <= n_nodes) {
        // Full tile (always taken for N % 16 == 0): unpredicated stores.
#pragma unroll
        for (int r = 0; r < 8; ++r)
            hbase[(size_t)(r + mhi) * DIM + nm] = acc[r];
    } else {
        // Ragged last tile: per-row guard.
#pragma unroll
        for (int r = 0; r < 8; ++r) {
            const int m = r + mhi;
            if (row0 + m < n_nodes) hbase[(size_t)m * DIM + nm] = acc[r];
        }
    }
}

// ---------------------------------------------------------------------------
// Kernel 2: out[dst[e]] += h[src[e]]  (segment-sum via L2-resident fp32
// atomics). Each wave grabs 32 edges at once (lane i loads edge base+i),
// then broadcasts each (src,dst) pair with __shfl (v_readlane) so the index
// fetch is done once per 32 edges instead of redundantly by all 32 lanes.
// Per edge: each lane moves a float4 (global_load_b128) from h[src] and
// issues 4 global_atomic_add_f32 into out[dst]. h (51 MB) and out (51 MB)
// both fit in the 192 MB L2, so the RMW traffic never leaves the cache.
// ---------------------------------------------------------------------------
__global__ void __launch_bounds__(256) gnn_scatter_kernel(
    const int* __restrict__ src, const int* __restrict__ dst,
    const float* __restrict__ h, float* __restrict__ out, int n_edges) {
    const int lane   = threadIdx.x & 31;
    const int wid    = (blockIdx.x * blockDim.x + threadIdx.x) >> 5;
    const int nwaves = (gridDim.x * blockDim.x) >> 5;

    for (int base = wid * 32; base < n_edges; base += nwaves * 32) {
        const int e = base + lane;
        int s = 0, d = 0;
        if (e < n_edges) {          // ragged tail: inactive lanes hold dummies
            s = src[e];
            d = dst[e];
        }
        const int cnt = min(32, n_edges - base);
        for (int j = 0; j < cnt; ++j) {
            const int sj = __shfl(s, j);    // uniform j -> v_readlane_b32
            const int dj = __shfl(d, j);
            const float4 v = ((const float4*)(h + (size_t)sj * DIM))[lane];
            float* o = out + (size_t)dj * DIM + lane * 4;
            unsafeAtomicAdd(o + 0, v.x);
            unsafeAtomicAdd(o + 1, v.y);
            unsafeAtomicAdd(o + 2, v.z);
            unsafeAtomicAdd(o + 3, v.w);
        }
    }
}

// ---------------------------------------------------------------------------
extern "C" void kernel_launch(void* const* d_in, const int* in_sizes, int n_in,
                              void* d_out, int out_size, void* d_ws, size_t ws_size,
                              hipStream_t stream) {
    const float* x  = (const float*)d_in[0];     // [N, 128] fp32
    const float* W  = (const float*)d_in[1];     // [128, 128] fp32
    const int*   ei = (const int*)d_in[2];       // [2, E] int32 (JAX default x64-off)
    float* out = (float*)d_out;                  // [N, 128] fp32
    float* h   = (float*)d_ws;                   // [N, 128] fp32 scratch (51.2 MB)

    const int n_nodes = in_sizes[0] / DIM;       // 100000
    const int n_edges = in_sizes[2] / 2;         // 1600000
    const int out_n4  = out_size / 4;

    // Zero the accumulator.
    gnn_zero_kernel<<<2048, 256, 0, stream>>>((float4*)out, out_n4);

    // h = x @ W^T  (16 rows per block, 8 waves cover the 128 out-channels)
    const int row_tiles = (n_nodes + 15) / 16;   // 6250
    gnn_gemm_wmma_kernel<<<row_tiles, 256, 0, stream>>>(x, W, h, n_nodes);

    // Edge scatter: 32 edges per wave iteration, grid-stride over 1.6M edges.
    gnn_scatter_kernel<<<4096, 256, 0, stream>>>(ei, ei + n_edges, h, out, n_edges);
}